// MatrixGraphConvolution_30494267802263
// MI455X (gfx1250) — compile-verified
//
#include <hip/hip_runtime.h>
#include <hip/hip_bf16.h>
#include <stdint.h>

// Problem constants (match reference).
#define NNODES 16384
#define NEDGES 524288
#define DD     128
#define MASKW  (NNODES / 32)   // 512 u32 words per adjacency row

typedef __attribute__((ext_vector_type(2))) float v2f;
typedef __attribute__((ext_vector_type(8))) float v8f;

// ---------------------------------------------------------------- zero fill
__global__ void mgc_zero_u32(unsigned int* p, int n) {
    int i = blockIdx.x * blockDim.x + threadIdx.x;
    if (i < n) p[i] = 0u;
}

// ------------------------------------------------- path A: bitmask (dedup)
// One bit per (dst, src) pair -> exact "set" semantics of the reference.
__global__ void mgc_edge_mask(const int* __restrict__ ei,
                              unsigned int* __restrict__ mask) {
    int e = blockIdx.x * blockDim.x + threadIdx.x;
    if (e >= NEDGES) return;
    int s = ei[e];            // edge_index[0][e]
    int d = ei[NEDGES + e];   // edge_index[1][e]
    atomicOr(&mask[(size_t)d * MASKW + (s >> 5)], 1u << (s & 31));
}

// One 128-thread block per node; thread t owns output channel t.
// Mask-row loads are uniform across the block (broadcast, L1/L2 friendly);
// x-row gathers are fully coalesced across channels.
__global__ void mgc_agg_from_mask(const unsigned int* __restrict__ mask,
                                  const float* __restrict__ x,
                                  float* __restrict__ agg) {
    int i = blockIdx.x;
    int t = threadIdx.x;
    const unsigned int* row = mask + (size_t)i * MASKW;
    float acc = 0.0f;
    int deg = 0;
    for (int w = 0; w < MASKW; ++w) {
        unsigned int m = row[w];
        deg += __popc(m);
        while (m) {
            int b = __ffs(m) - 1;
            m &= m - 1;
            int j = (w << 5) + b;
            acc += x[(size_t)j * DD + t];
        }
    }
    float dinv = 1.0f / fmaxf((float)deg, 1.0f);
    agg[(size_t)i * DD + t] = acc * dinv;   // D^-1 * (A @ x), pre-scaled
}

// -------------------------------- path B: atomic fallback (small ws_size)
// Duplicate edges are counted instead of collapsed (rare for random edges).
__global__ void mgc_edge_atomic(const int* __restrict__ ei,
                                const float* __restrict__ x,
                                float* __restrict__ agg,
                                float* __restrict__ deg) {
    long long gid = (long long)blockIdx.x * blockDim.x + threadIdx.x;
    if (gid >= (long long)NEDGES * DD) return;
    int e = (int)(gid >> 7);
    int t = (int)(gid & (DD - 1));
    int s = ei[e];
    int d = ei[NEDGES + e];
    atomicAdd(&agg[(size_t)d * DD + t], x[(size_t)s * DD + t]);
    if (t == 0) atomicAdd(&deg[d], 1.0f);
}

__global__ void mgc_scale_agg(float* __restrict__ agg,
                              const float* __restrict__ deg) {
    int i = blockIdx.x;
    int t = threadIdx.x;
    agg[(size_t)i * DD + t] *= 1.0f / fmaxf(deg[i], 1.0f);
}

// ------------------------------------------------------------- WMMA GEMM
// out = agg @ W^T + x @ B^T  ==  [agg | x](16384x256) @ [W | B]^T
// One wave per 16x16 output tile; K swept in steps of 4 with
// V_WMMA_F32_16X16X4_F32 (full f32 precision, matches reference math).
//
// Per-lane operand layout (ISA 7.12.2, 32-bit A 16x4 / B 4x16 / C 16x16):
//   h = lane>>4, ml = lane&15
//   A vgpr{0,1} = A[m = ml][k + 2h + {0,1}]          -> contiguous float2
//   B vgpr{0,1} = Bmat[k + 2h + {0,1}][n = ml]       -> Wcat[n][k+2h+{0,1}]
//   C vgpr r    = C[m = r + 8h][n = ml]
__global__ __launch_bounds__(256)
void mgc_gemm_wmma(const float* __restrict__ agg,
                   const float* __restrict__ x,
                   const float* __restrict__ W,
                   const float* __restrict__ B,
                   float* __restrict__ out) {
    int lane  = threadIdx.x & 31;
    int wave  = threadIdx.x >> 5;
    int tile  = blockIdx.x * 8 + wave;     // 8192 tiles total
    int mtile = tile >> 3;                 // 1024 M-tiles
    int ntile = tile & 7;                  // 8 N-tiles
    int h  = lane >> 4;
    int ml = lane & 15;

    const float* arowA = agg + (size_t)(mtile * 16 + ml) * DD;
    const float* arowX = x   + (size_t)(mtile * 16 + ml) * DD;
    const float* browW = W   + (size_t)(ntile * 16 + ml) * DD;
    const float* browB = B   + (size_t)(ntile * 16 + ml) * DD;

    v8f c = {};
    // K = 0..127: agg x W^T
#pragma unroll 8
    for (int k = 0; k < DD; k += 4) {
        int kk = k + 2 * h;                         // even -> 8B aligned
        v2f a = *(const v2f*)(arowA + kk);
        v2f b = *(const v2f*)(browW + kk);
        c = __builtin_amdgcn_wmma_f32_16x16x4_f32(
                false, a, false, b, (short)0, c, false, false);
    }
    // K = 128..255: x x B^T
#pragma unroll 8
    for (int k = 0; k < DD; k += 4) {
        int kk = k + 2 * h;
        v2f a = *(const v2f*)(arowX + kk);
        v2f b = *(const v2f*)(browB + kk);
        c = __builtin_amdgcn_wmma_f32_16x16x4_f32(
                false, a, false, b, (short)0, c, false, false);
    }

    float* orow = out + (size_t)(mtile * 16 + h * 8) * DD + ntile * 16 + ml;
#pragma unroll
    for (int r = 0; r < 8; ++r)
        orow[(size_t)r * DD] = c[r];
}

// ------------------------------------------------------------------ host
extern "C" void kernel_launch(void* const* d_in, const int* in_sizes, int n_in,
                              void* d_out, int out_size, void* d_ws, size_t ws_size,
                              hipStream_t stream) {
    const float* x  = (const float*)d_in[0];   // (16384, 128) f32
    const int*   ei = (const int*)d_in[1];     // (2, 524288) int
    const float* W  = (const float*)d_in[2];   // (128, 128) f32
    const float* B  = (const float*)d_in[3];   // (128, 128) f32
    float* out = (float*)d_out;

    uint8_t* ws = (uint8_t*)d_ws;
    const size_t aggBytes = (size_t)NNODES * DD * sizeof(float);   // 8 MB
    float*        agg  = (float*)ws;
    float*        deg  = (float*)(ws + aggBytes);                  // 64 KB
    unsigned int* mask = (unsigned int*)(ws + aggBytes + (1u << 20)); // 32 MB
    const size_t needMask =
        aggBytes + (1u << 20) + (size_t)NNODES * MASKW * sizeof(unsigned int);

    if (ws_size >= needMask) {
        // Exact dedup path (matches the reference's set semantics).
        int nm = NNODES * MASKW;  // 8.4M words
        mgc_zero_u32<<<(nm + 255) / 256, 256, 0, stream>>>(mask, nm);
        mgc_edge_mask<<<(NEDGES + 255) / 256, 256, 0, stream>>>(ei, mask);
        mgc_agg_from_mask<<<NNODES, DD, 0, stream>>>(mask, x, agg);
    } else {
        // Atomic fallback (duplicates counted).
        int na = NNODES * DD;
        mgc_zero_u32<<<(na + 255) / 256, 256, 0, stream>>>((unsigned int*)agg, na);
        mgc_zero_u32<<<(NNODES + 255) / 256, 256, 0, stream>>>((unsigned int*)deg, NNODES);
        long long ne = (long long)NEDGES * DD;
        mgc_edge_atomic<<<(int)((ne + 255) / 256), 256, 0, stream>>>(ei, x, agg, deg);
        mgc_scale_agg<<<NNODES, DD, 0, stream>>>(agg, deg);
    }

    // 8192 tiles / 8 waves per 256-thread block -> 1024 blocks.
    mgc_gemm_wmma<<<1024, 256, 0, stream>>>(agg, x, W, B, out);
}